// vlad_core_euc_45775761440891
// MI455X (gfx1250) — compile-verified
//
#include <hip/hip_runtime.h>
#include <math.h>

typedef float v2f __attribute__((ext_vector_type(2)));
typedef float v8f __attribute__((ext_vector_type(8)));
typedef __attribute__((address_space(3))) float lds_f;

#define ALPHA_F 10.0f
#define EPS_F   1e-12f

constexpr int Bn = 32;
constexpr int Nn = 16384;
constexpr int Dn = 64;
constexpr int Cn = 32;
constexpr int Sn = 8;                                   // N-slices per batch
constexpr int ROWS_PER_SLICE  = Nn / Sn;                // 2048
constexpr int TILES_PER_SLICE = ROWS_PER_SLICE / 16;    // 128
constexpr int WAVES = 4;                                // 128 threads/block
constexpr int PART_STRIDE = Dn * Cn + Cn;               // 2080 floats per (b,s)
constexpr int XT_STRIDE = 68;   // padded x-tile row stride: 16B-aligned rows, conflict-free A reads
constexpr int AP_STRIDE = 18;   // padded assign column stride: even -> 8B-aligned v2f pairs

static __device__ __forceinline__ v8f wmma_f32_16x16x4(v2f a, v2f b, v8f c) {
  // D = A(16x4, f32) * B(4x16, f32) + C(16x16, f32), wave32
  return __builtin_amdgcn_wmma_f32_16x16x4_f32(false, a, false, b, (short)0, c, false, false);
}

__global__ __launch_bounds__(WAVES * 32)
void vlad_pass1_45775761440891(const float* __restrict__ x,
                               const float* __restrict__ cent,
                               float* __restrict__ ws)
{
  // pre-packed GEMM1 B fragments: bfrag[j][ct][lane] = (cent[k][c], cent[k+1][c])
  __shared__ __align__(16) float2 bfrag_s[16 * 2 * 32];               // 8 KB
  __shared__ float clen_s[Cn];
  __shared__ float xsum_s[Dn * Cn];                                   // 8 KB
  __shared__ float cwt_s[Cn];
  __shared__ __align__(16) float xtile_s[WAVES][2][16 * XT_STRIDE];   // 34 KB, double-buffered
  __shared__ __align__(16) float apk_s[WAVES][Cn * AP_STRIDE];        // 9 KB

  const int b   = blockIdx.x / Sn;
  const int s   = blockIdx.x % Sn;
  const int tid = threadIdx.x;
  const int wv  = tid >> 5;
  const int ln  = tid & 31;
  const int half = ln >> 4;     // 0: lanes 0-15, 1: lanes 16-31
  const int lr   = ln & 15;

  // build packed centroid B-fragments + zero accumulators (once per block)
  for (int e = tid; e < 16 * 2 * 32; e += WAVES * 32) {
    const int j  = e >> 6;
    const int ct = (e >> 5) & 1;
    const int l  = e & 31;
    const int k  = 4 * j + 2 * (l >> 4);
    const int c  = ct * 16 + (l & 15);
    bfrag_s[e] = make_float2(cent[k * Cn + c], cent[(k + 1) * Cn + c]);
  }
  for (int i = tid; i < Dn * Cn; i += WAVES * 32) xsum_s[i] = 0.0f;
  if (tid < Cn) {
    cwt_s[tid] = 0.0f;
    float ssum = 0.0f;
    for (int d = 0; d < Dn; ++d) { float v = cent[d * Cn + tid]; ssum += v * v; }
    clen_s[tid] = ssum;
  }
  __syncthreads();

  const float aclen_lo = ALPHA_F * clen_s[lr];
  const float aclen_hi = ALPHA_F * clen_s[lr + 16];
  float* ap = apk_s[wv];
  const v2f* bf = (const v2f*)bfrag_s;

  // persistent x_sum accumulators: [dtile 0..3][ctile 0..1] of 16x16 f32 frags
  v8f xs[4][2];
  #pragma unroll
  for (int dt = 0; dt < 4; ++dt)
    #pragma unroll
    for (int ct = 0; ct < 2; ++ct)
      #pragma unroll
      for (int r = 0; r < 8; ++r) xs[dt][ct][r] = 0.0f;
  float cw0 = 0.0f, cw1 = 0.0f;

  const float* xb = x + (size_t)b * Nn * Dn + (size_t)s * ROWS_PER_SLICE * Dn;

  // async copy of one 16x64 f32 tile: global -> LDS (padded rows), 16B per lane chunk
  auto issue_tile = [&](const float* xg, float* dst) {
    #pragma unroll
    for (int i = 0; i < 8; ++i) {
      const int idx = i * 32 + ln;
      const int row = idx >> 4, c4 = idx & 15;
      lds_f* lp = (lds_f*)(dst + row * XT_STRIDE + c4 * 4);
      const float* gp = xg + row * Dn + c4 * 4;
      asm volatile("global_load_async_to_lds_b128 %0, %1, off"
                   :: "v"(lp), "v"(gp) : "memory");
    }
  };

  // prologue: start first tile
  issue_tile(xb + (size_t)wv * 16 * Dn, xtile_s[wv][0]);

  int pb = 0;
  for (int t = wv; t < TILES_PER_SLICE; t += WAVES, pb ^= 1) {
    // kick off next tile into the alternate buffer, then wait for current
    if (t + WAVES < TILES_PER_SLICE) {
      asm volatile("s_wait_dscnt 0" ::: "memory");      // alt-buffer reuse guard
      issue_tile(xb + (size_t)(t + WAVES) * 16 * Dn, xtile_s[wv][pb ^ 1]);
      asm volatile("s_wait_asynccnt 8" ::: "memory");   // oldest 8 (current tile) done
    } else {
      asm volatile("s_wait_asynccnt 0" ::: "memory");
    }
    float* xt = xtile_s[wv][pb];

    // ---- A-fragment preload (raw x) + row sum-of-squares in one pass ----
    v2f axf[16];
    float ssq = 0.0f;
    #pragma unroll
    for (int j = 0; j < 16; ++j) {
      axf[j] = *(const v2f*)(xt + lr * XT_STRIDE + 4 * j + 2 * half);
      ssq += axf[j].x * axf[j].x + axf[j].y * axf[j].y;
    }
    ssq += __shfl_xor(ssq, 16, 32);                      // both halves = full row
    const float rn = 1.0f / fmaxf(sqrtf(ssq), EPS_F);    // 1/max(||x_row||, eps)

    // ---- GEMM1 on raw x: dot_raw(16x32) = x(16x64) * cent(64x32) ----
    v8f d0, d1;
    #pragma unroll
    for (int r = 0; r < 8; ++r) { d0[r] = 0.0f; d1[r] = 0.0f; }
    #pragma unroll
    for (int j = 0; j < 16; ++j) {
      v2f b0 = bf[(j * 2 + 0) * 32 + ln];
      v2f b1 = bf[(j * 2 + 1) * 32 + ln];
      d0 = wmma_f32_16x16x4(axf[j], b0, d0);
      d1 = wmma_f32_16x16x4(axf[j], b1, d1);
    }

    // ---- softmax over C=32 per row; logits bounded (unit rows/cols) so no
    //      max-subtraction needed; fold rn into logits and into assign rows ----
    #pragma unroll
    for (int r = 0; r < 8; ++r) {
      const float rm = __shfl(rn, r + 8 * half, 32);     // rn of D-frag row m
      const float ar = 2.0f * ALPHA_F * rm;
      float e0 = __expf(ar * d0[r] - aclen_lo);
      float e1 = __expf(ar * d1[r] - aclen_hi);
      float sm = e0 + e1;
      sm += __shfl_xor(sm, 1, 32);
      sm += __shfl_xor(sm, 2, 32);
      sm += __shfl_xor(sm, 4, 32);
      sm += __shfl_xor(sm, 8, 32);
      const float inv = 1.0f / sm;
      e0 *= inv; e1 *= inv;                              // assign (unscaled)
      const int m = r + 8 * half;                        // D-frag row id
      ap[lr * AP_STRIDE + m]        = e0 * rm;           // rn-folded for GEMM2
      ap[(lr + 16) * AP_STRIDE + m] = e1 * rm;
      cw0 += e0; cw1 += e1;                              // column-sum partials
    }
    asm volatile("s_wait_dscnt 0" ::: "memory");

    // ---- GEMM2: x_sum(64x32) += x_raw^T(64x16) * (rn*assign)(16x32) ----
    #pragma unroll
    for (int kc = 0; kc < 4; ++kc) {
      const int kr = 4 * kc + 2 * half;                  // this lane's row pair
      v2f bb0 = *(const v2f*)(ap + lr * AP_STRIDE + kr);
      v2f bb1 = *(const v2f*)(ap + (lr + 16) * AP_STRIDE + kr);
      #pragma unroll
      for (int dt = 0; dt < 4; ++dt) {
        v2f a2;                                          // A2: M=d, K=row pair
        a2.x = xt[kr * XT_STRIDE + dt * 16 + lr];
        a2.y = xt[(kr + 1) * XT_STRIDE + dt * 16 + lr];
        xs[dt][0] = wmma_f32_16x16x4(a2, bb0, xs[dt][0]);
        xs[dt][1] = wmma_f32_16x16x4(a2, bb1, xs[dt][1]);
      }
    }
  }

  // ---- cross-wave reduction via LDS float atomics (ds_add_f32) ----
  #pragma unroll
  for (int dt = 0; dt < 4; ++dt)
    #pragma unroll
    for (int ct = 0; ct < 2; ++ct)
      #pragma unroll
      for (int r = 0; r < 8; ++r) {
        const int d = dt * 16 + r + 8 * half;
        const int c = ct * 16 + lr;
        atomicAdd(&xsum_s[d * Cn + c], xs[dt][ct][r]);
      }
  cw0 += __shfl_xor(cw0, 16, 32);
  cw1 += __shfl_xor(cw1, 16, 32);
  if (ln < 16) {
    atomicAdd(&cwt_s[lr], cw0);
    atomicAdd(&cwt_s[lr + 16], cw1);
  }
  __syncthreads();

  float* wsp = ws + (size_t)(b * Sn + s) * PART_STRIDE;
  for (int i = tid; i < Dn * Cn; i += WAVES * 32) wsp[i] = xsum_s[i];
  if (tid < Cn) wsp[Dn * Cn + tid] = cwt_s[tid];
}

__global__ __launch_bounds__(256)
void vlad_pass2_45775761440891(const float* __restrict__ cent,
                               const float* __restrict__ ws,
                               float* __restrict__ out)
{
  __shared__ float cw_s[Cn];
  __shared__ float gsum;
  const int b = blockIdx.x;
  const int tid = threadIdx.x;
  if (tid == 0) gsum = 0.0f;
  const float* wsb = ws + (size_t)b * Sn * PART_STRIDE;
  if (tid < Cn) {
    float acc = 0.0f;
    for (int sl = 0; sl < Sn; ++sl) acc += wsb[sl * PART_STRIDE + Dn * Cn + tid];
    cw_s[tid] = acc;
  }
  __syncthreads();

  const int c = tid >> 3;            // 8 threads per cluster row (8 d's each)
  const int o0 = tid * 8;
  float vals[8];
  float ss = 0.0f;
  #pragma unroll
  for (int i = 0; i < 8; ++i) {
    const int o = o0 + i;
    const int d = o & (Dn - 1);
    const int e = d * Cn + c;
    float v = 0.0f;
    for (int sl = 0; sl < Sn; ++sl) v += wsb[sl * PART_STRIDE + e];
    v -= cent[e] * cw_s[c];          // residual vs. weighted centroid
    vals[i] = v;
    ss += v * v;
  }
  // intra-normalize each cluster row (64 d's -> 8 threads, aligned groups)
  ss += __shfl_xor(ss, 1, 32);
  ss += __shfl_xor(ss, 2, 32);
  ss += __shfl_xor(ss, 4, 32);
  const float rnc = 1.0f / fmaxf(sqrtf(ss), EPS_F);
  float ps = 0.0f;
  #pragma unroll
  for (int i = 0; i < 8; ++i) { vals[i] *= rnc; ps += vals[i] * vals[i]; }
  atomicAdd(&gsum, ps);
  __syncthreads();
  const float g = 1.0f / fmaxf(sqrtf(gsum), EPS_F);
  #pragma unroll
  for (int i = 0; i < 8; ++i) out[(size_t)b * (Dn * Cn) + o0 + i] = vals[i] * g;
}

extern "C" void kernel_launch(void* const* d_in, const int* in_sizes, int n_in,
                              void* d_out, int out_size, void* d_ws, size_t ws_size,
                              hipStream_t stream) {
  (void)in_sizes; (void)n_in; (void)out_size; (void)ws_size;
  const float* x    = (const float*)d_in[0];   // [B, N, D] f32
  const float* cent = (const float*)d_in[1];   // [D, C]    f32
  float* out = (float*)d_out;                  // [B, C*D]  f32
  float* ws  = (float*)d_ws;                   // >= B*Sn*2080 floats (~2.03 MB)

  vlad_pass1_45775761440891<<<Bn * Sn, WAVES * 32, 0, stream>>>(x, cent, ws);
  vlad_pass2_45775761440891<<<Bn, 256, 0, stream>>>(cent, ws, out);
}